// SubwordEmbedding_64158221468108
// MI455X (gfx1250) — compile-verified
//
#include <hip/hip_runtime.h>
#include <hip/hip_bf16.h>
#include <stdint.h>

// SubwordEmbedding: out[b,s,:] = word_table[input_ids[b,s],:] +
//                   (sum_k mask[b,s,k]*sub_table[subword_ids[b,s,k],:]) / max(sum_k mask,1)
// B=32, S=512 -> P=16384 positions, K=16, D=128. Pure HBM gather workload:
// ~85 MB after mask-skipping -> ~3.7 us at 23.3 TB/s. No matrix math exists here;
// the CDNA5 path that matters is async global->LDS gathers (ASYNCcnt-tracked),
// which keep 512 B row requests in flight with zero destination-VGPR cost.

#define KSUB 16
#define DDIM 128
#define WAVES_PER_BLOCK 8
#define BLOCK_THREADS (WAVES_PER_BLOCK * 32)

#if defined(__gfx1250__) && __has_builtin(__builtin_amdgcn_global_load_async_to_lds_b128)
#define USE_ASYNC_LDS 1
#else
#define USE_ASYNC_LDS 0
#endif

typedef float f4 __attribute__((ext_vector_type(4)));

#if USE_ASYNC_LDS
// Builtin signature (verified by hipcc diagnostics): (v4i addrspace(1)*,
// v4i addrspace(3)*, imm offset, imm cpol).
typedef int v4i __attribute__((ext_vector_type(4)));
typedef __attribute__((address_space(1))) v4i* g_v4i_ptr;
typedef __attribute__((address_space(3))) v4i* l_v4i_ptr;
#endif

__global__ __launch_bounds__(BLOCK_THREADS)
void subword_embed_kernel(const int* __restrict__ input_ids,
                          const int* __restrict__ subword_ids,
                          const int* __restrict__ subword_mask,
                          const float* __restrict__ word_table,
                          const float* __restrict__ sub_table,
                          float* __restrict__ out,
                          int P)
{
#if USE_ASYNC_LDS
    __shared__ __align__(16) float stage[WAVES_PER_BLOCK][KSUB * DDIM]; // 64 KB/WG
#endif
    const int lane = threadIdx.x & 31;       // wave32
    const int wv   = threadIdx.x >> 5;
    const int pos  = blockIdx.x * WAVES_PER_BLOCK + wv;     // uniform per wave
    if (pos >= P) return;                                   // wave-uniform exit

    // One predicated load covers both arrays: lanes 0..15 -> the 16 subword
    // ids, lanes 16..31 -> the 16 mask words. Ballot bits [31:16] are then
    // exactly the wave-uniform k-mask (SGPR -> scalar branches per k).
    const int  kidx = lane & (KSUB - 1);
    const int* src  = (lane < KSUB)
                        ? (subword_ids  + (size_t)pos * KSUB + kidx)
                        : (subword_mask + (size_t)pos * KSUB + kidx);
    const int x = *src;                       // id in lanes 0..15, mask in 16..31

    const unsigned long long bal = __ballot(x != 0);
    const unsigned mm  = (unsigned)(bal >> KSUB) & 0xFFFFu;   // mask[k] != 0
    const int      cnt = __popc(mm);
    const float    inv = 1.0f / (float)(cnt > 1 ? cnt : 1);

    f4 acc = {0.f, 0.f, 0.f, 0.f};

#if USE_ASYNC_LDS
    // Issue all masked row gathers HBM -> LDS asynchronously: one 512 B b128
    // per row per wave (lane l moves bytes [16l,16l+16)).
    float* rows = stage[wv];
    #pragma unroll
    for (int k = 0; k < KSUB; ++k) {
        if ((mm >> k) & 1u) {
            const int r = __shfl(x, k, 32);   // id k lives in lane k
            g_v4i_ptr g = (g_v4i_ptr)(sub_table + (size_t)r * DDIM + lane * 4);
            l_v4i_ptr l = (l_v4i_ptr)(rows + k * DDIM + lane * 4);
            __builtin_amdgcn_global_load_async_to_lds_b128(g, l, 0, 0);
        }
    }
#endif

    // Word-table row (uniform id -> scalar base), issued while the async
    // gathers are in flight. 512 B coalesced per wave.
    const int wid = __builtin_amdgcn_readfirstlane(input_ids[pos]);
    const f4 w = ((const f4*)(word_table + (size_t)wid * DDIM))[lane];

#if USE_ASYNC_LDS
    #if __has_builtin(__builtin_amdgcn_s_wait_asynccnt)
    __builtin_amdgcn_s_wait_asynccnt(0);
    #else
    asm volatile("s_wait_asynccnt 0" ::: "memory");
    #endif
    #pragma unroll
    for (int k = 0; k < KSUB; ++k) {
        if ((mm >> k) & 1u) {
            const f4 v = *(const f4*)(rows + k * DDIM + lane * 4);
            acc += v;                          // packed f32 adds
        }
    }
#else
    // Fallback: direct loads, issued in batches of 8 rows (two-pass so all
    // loads in a batch are outstanding before the first consumer wait).
    #pragma unroll
    for (int kk = 0; kk < KSUB; kk += 8) {
        f4 v[8];
        #pragma unroll
        for (int j = 0; j < 8; ++j) {
            v[j] = (f4){0.f, 0.f, 0.f, 0.f};
            if ((mm >> (kk + j)) & 1u) {
                const int r = __shfl(x, kk + j, 32);
                v[j] = ((const f4*)(sub_table + (size_t)r * DDIM))[lane];
            }
        }
        #pragma unroll
        for (int j = 0; j < 8; ++j) acc += v[j];
    }
#endif

    const f4 invv = {inv, inv, inv, inv};
    const f4 o = acc * invv + w;               // packed fma
    ((f4*)(out + (size_t)pos * DDIM))[lane] = o;   // 512 B coalesced store

    (void)subword_mask;  // used via predicated pointer select above
}

extern "C" void kernel_launch(void* const* d_in, const int* in_sizes, int n_in,
                              void* d_out, int out_size, void* d_ws, size_t ws_size,
                              hipStream_t stream) {
    const int*   input_ids    = (const int*)d_in[0];   // [B*S] int32
    const int*   subword_ids  = (const int*)d_in[1];   // [B*S*K] int32
    const int*   subword_mask = (const int*)d_in[2];   // [B*S*K] int32
    const float* word_table   = (const float*)d_in[3]; // [VOCAB*D] f32
    const float* sub_table    = (const float*)d_in[4]; // [BUCKET*D] f32
    float*       out          = (float*)d_out;         // [B*S*D] f32

    const int P = in_sizes[0];                         // B*S positions
    const int blocks = (P + WAVES_PER_BLOCK - 1) / WAVES_PER_BLOCK;
    subword_embed_kernel<<<blocks, BLOCK_THREADS, 0, stream>>>(
        input_ids, subword_ids, subword_mask, word_table, sub_table, out, P);
}